// DAWNBlock_64278480552599
// MI455X (gfx1250) — compile-verified
//
#include <hip/hip_runtime.h>
#include <math.h>

#define NF 2048
#define NR 1024
#define NV 1536
#define NUSE 4608          // NF+NR+NV (last NK=1024 logits are never used)
#define B_ 8
#define S_ 2048
#define D_ 2048
#define DS_ 64
#define ROWS (B_*S_)       // 16384
#define NTILES (ROWS/16)   // 1024 row-tiles of 16

typedef float v2f __attribute__((ext_vector_type(2)));
typedef float v8f __attribute__((ext_vector_type(8)));
typedef int   v4i __attribute__((ext_vector_type(4)));

// pointer-to-v4i in specific address spaces (param types of the async builtin)
typedef __attribute__((address_space(1))) v4i* as1_v4i_p;   // global
typedef __attribute__((address_space(3))) v4i* as3_v4i_p;   // LDS

// ---- CDNA5 async Global->LDS path (guarded: falls back to plain copy) -----
#if defined(__has_builtin)
# if __has_builtin(__builtin_amdgcn_global_load_async_to_lds_b128) && \
     __has_builtin(__builtin_amdgcn_s_wait_asynccnt)
#  define HAVE_ASYNC_LDS 1
# endif
#endif
#ifndef HAVE_ASYNC_LDS
# define HAVE_ASYNC_LDS 0
#endif

__device__ __forceinline__ void async_copy_b128(const float* g, float* l) {
#if HAVE_ASYNC_LDS
  __builtin_amdgcn_global_load_async_to_lds_b128(
      (as1_v4i_p)(void*)g, (as3_v4i_p)(void*)l, 0, 0);
#else
  // fallback: 16-byte copy through VGPRs
  *(float4*)l = *(const float4*)g;
#endif
}

__device__ __forceinline__ void async_wait0() {
#if HAVE_ASYNC_LDS
  __builtin_amdgcn_s_wait_asynccnt(0);
#endif
}

__device__ __forceinline__ v8f wmma4(v2f a, v2f b, v8f c) {
  // D = A(16x4 f32) * B(4x16 f32) + C(16x16 f32)
  return __builtin_amdgcn_wmma_f32_16x16x4_f32(false, a, false, b, (short)0, c,
                                               false, false);
}

__device__ __forceinline__ v8f zero8() {
  v8f z = {0.f,0.f,0.f,0.f,0.f,0.f,0.f,0.f};
  return z;
}

// ---------------------------------------------------------------------------
// Prep: normalize neuron_emb rows (first NUSE only) + excitability bias
// exc(u) = clamp(1 - u/TAU, 0, 1) * EXC_W, TAU = EXC_W = 1
// ---------------------------------------------------------------------------
__global__ void k_prep(const float* __restrict__ emb,
                       const float* __restrict__ uf,
                       const float* __restrict__ ur,
                       const float* __restrict__ uv,
                       float* __restrict__ embn,
                       float* __restrict__ bias) {
  int n = blockIdx.x * blockDim.x + threadIdx.x;
  if (n >= NUSE) return;
  const float* row = emb + (size_t)n * DS_;
  float acc = 0.f;
  #pragma unroll 4
  for (int j = 0; j < DS_; ++j) { float v = row[j]; acc += v * v; }
  float inv = 1.0f / (sqrtf(acc) + 1e-12f);
  float* orow = embn + (size_t)n * DS_;
  #pragma unroll 4
  for (int j = 0; j < DS_; ++j) orow[j] = row[j] * inv;

  float u = (n < NF) ? uf[n] : (n < NF + NR) ? ur[n - NF] : uv[n - NF - NR];
  float e = 1.0f - u;
  bias[n] = fminf(fmaxf(e, 0.0f), 1.0f);
}

// ---------------------------------------------------------------------------
// Pack W into WMMA-B pair layout: Wp[kp*64+n] = {W[2kp][n], W[2kp+1][n]}.
// For any 128-row K-chunk this layout is linear (float offset kc*64), so the
// async b128 DMA addressing is identical — but the per-lane B fragment
// becomes a single aligned ds_load_b64 (no register re-packing movs).
// ---------------------------------------------------------------------------
__global__ void k_packW(const float* __restrict__ W, float* __restrict__ Wp) {
  int p = blockIdx.x * blockDim.x + threadIdx.x;      // pair index
  if (p >= (D_ / 2) * 64) return;
  int kp = p >> 6, n = p & 63;
  v2f v;
  v.x = W[(size_t)(2 * kp    ) * 64 + n];
  v.y = W[(size_t)(2 * kp + 1) * 64 + n];
  *(v2f*)(Wp + (size_t)p * 2) = v;
}

// ---------------------------------------------------------------------------
// Projection GEMM: h[16384,64] = x[16384,2048] @ W[2048,64] + b
// 4 waves/block, each wave: 16 rows x 64 cols via 4 accumulators.
// Pair-packed W staged into LDS per 128-K chunk via async DMA (ASYNCcnt);
// A fragments streamed from global (x read exactly once, 128 MB @ 23.3 TB/s).
// ---------------------------------------------------------------------------
__global__ void __launch_bounds__(128)
k_proj(const float* __restrict__ x, const float* __restrict__ Wp,
       const float* __restrict__ bproj, float* __restrict__ h) {
  __shared__ __attribute__((aligned(16))) v2f Wt2[64 * 64];  // 32 KB
  const int tid  = threadIdx.x;
  const int wave = tid >> 5, lane = tid & 31;
  const int nl   = lane & 15, hi = lane >> 4;
  const int koff = hi * 2;
  const int row0 = blockIdx.x * 64 + wave * 16;

  v8f acc[4];
  #pragma unroll
  for (int t = 0; t < 4; ++t) acc[t] = zero8();

  for (int kc = 0; kc < D_; kc += 128) {
    __syncthreads();                    // prior readers done before DMA lands
    #pragma unroll
    for (int i = 0; i < 4; ++i)         // 128 thr * 4 * 16B = 8 KB, 16B per 64B of chunk
      async_copy_b128(Wp + (size_t)kc * 64 + i * 2048 + tid * 16,
                      (float*)Wt2 + i * 2048 + tid * 16);
    async_wait0();                      // this wave's DMA complete
    __syncthreads();                    // all waves' DMA complete

    const float* xrow = x + (size_t)(row0 + nl) * D_ + kc + koff;
    __builtin_prefetch(xrow + 128, 0, 1);   // next chunk of the x stream
    #pragma unroll 4
    for (int k2 = 0; k2 < 128; k2 += 4) {
      v2f a = *(const v2f*)(xrow + k2);
      const int kp = (k2 >> 1) + hi;    // pair row for this lane's K half
      #pragma unroll
      for (int t = 0; t < 4; ++t)
        acc[t] = wmma4(a, Wt2[kp * 64 + t * 16 + nl], acc[t]);
    }
  }
  #pragma unroll
  for (int t = 0; t < 4; ++t) {
    int n = t * 16 + nl;
    float bn = bproj[n];
    #pragma unroll
    for (int r = 0; r < 8; ++r) {
      int row = row0 + r + hi * 8;     // C layout: lanes>=16 hold M=r+8
      h[(size_t)row * 64 + n] = acc[t][r] + bn;
    }
  }
}

// ---------------------------------------------------------------------------
// Fused logits + bias + segment softmax + reductions.
// Grid (NTILES, 3): one block = 16 rows x one full segment.
// Logits tile lives entirely in LDS (max 16x2048x4B = 128KB of 320KB/WGP);
// emits per-tile partial sums (deterministic, no atomics).
// ---------------------------------------------------------------------------
__global__ void __launch_bounds__(256)
k_logits(const float* __restrict__ h, const float* __restrict__ embn,
         const float* __restrict__ bias, const float* __restrict__ importance,
         float* __restrict__ partA, float* __restrict__ partB) {
  __shared__ __attribute__((aligned(16))) float hS[16 * 64];
  __shared__ float logitsS[16 * 2048];   // 128 KB, sized for largest segment
  __shared__ float red[256];
  __shared__ float invZ[16], impS[16];

  const int seg    = blockIdx.y;
  const int segN   = (seg == 0) ? NF : (seg == 1) ? NR : NV;
  const int segOff = (seg == 0) ? 0  : (seg == 1) ? NF : NF + NR;
  const int tile   = blockIdx.x;
  const int row0   = tile * 16;
  const int tid    = threadIdx.x;

  // stage 16x64 h tile: exactly one b128 async DMA per thread
  async_copy_b128(h + (size_t)row0 * 64 + tid * 4, hS + tid * 4);
  async_wait0();
  __syncthreads();

  const int wave = tid >> 5, lane = tid & 31;
  const int nl = lane & 15, hi2 = lane >> 4, koff = hi2 * 2;

  // hoist all 16 A k-fragments (K=64) for these 16 rows into registers
  v2f af[16];
  #pragma unroll
  for (int kk = 0; kk < 16; ++kk)
    af[kk] = *(const v2f*)&hS[nl * 64 + kk * 4 + koff];

  const int nt = segN >> 7;              // 8 waves * 16 cols = 128 cols/step
  for (int t = 0; t < nt; ++t) {
    const int col = t * 128 + wave * 16;
    const float* erow = embn + (size_t)(segOff + col + nl) * 64 + koff;
    v8f c = zero8();
    #pragma unroll
    for (int kk = 0; kk < 16; ++kk) {
      v2f b = *(const v2f*)(erow + kk * 4);
      c = wmma4(af[kk], b, c);
    }
    float bn = bias[segOff + col + nl];
    #pragma unroll
    for (int r = 0; r < 8; ++r) {
      int m = r + hi2 * 8;
      logitsS[m * segN + col + nl] = c[r] + bn;
    }
  }
  __syncthreads();

  // segment softmax: 16 rows x 16 threads each
  const int r = tid >> 4, li = tid & 15;
  float mx = -INFINITY;
  for (int n = li; n < segN; n += 16) mx = fmaxf(mx, logitsS[r * segN + n]);
  red[tid] = mx; __syncthreads();
  for (int s = 8; s > 0; s >>= 1) {
    if (li < s) red[tid] = fmaxf(red[tid], red[tid + s]);
    __syncthreads();
  }
  const float rowmax = red[r << 4];
  __syncthreads();

  float ssum = 0.f;
  for (int n = li; n < segN; n += 16) {
    float p = __expf(logitsS[r * segN + n] - rowmax);
    logitsS[r * segN + n] = p;
    ssum += p;
  }
  red[tid] = ssum; __syncthreads();
  for (int s = 8; s > 0; s >>= 1) {
    if (li < s) red[tid] += red[tid + s];
    __syncthreads();
  }
  if (li == 0) {
    invZ[r] = 1.0f / red[r << 4];
    impS[r] = importance[row0 + r];     // rows of a tile share batch index
  }
  __syncthreads();

  // per-neuron reductions over the 16 rows of this tile
  for (int n = tid; n < segN; n += 256) {
    float s1 = 0.f, s2 = 0.f;
    #pragma unroll
    for (int rr = 0; rr < 16; ++rr) {
      float p = logitsS[rr * segN + n] * invZ[rr];
      s1 += p;
      s2 += impS[rr] * p;
    }
    partA[(size_t)tile * NUSE + segOff + n] = s1;   // pref sum (aux loss)
    partB[(size_t)tile * NUSE + segOff + n] = s2;   // importance-weighted pool
  }
}

// ---------------------------------------------------------------------------
// Deterministic tile reduction: dense[b,n] and aux pref-sum[n]
// ---------------------------------------------------------------------------
__global__ void k_reduce(const float* __restrict__ partA,
                         const float* __restrict__ partB,
                         float* __restrict__ auxsum,
                         float* __restrict__ dense) {
  int idx = blockIdx.x * blockDim.x + threadIdx.x;
  if (idx < B_ * NUSE) {
    int b = idx / NUSE, n = idx - b * NUSE;
    const float* p = partB + (size_t)b * (S_ / 16) * NUSE + n;
    float s = 0.f;
    for (int t = 0; t < S_ / 16; ++t) s += p[(size_t)t * NUSE];
    dense[idx] = s;
  }
  if (idx < NUSE) {
    const float* p = partA + idx;
    float s = 0.f;
    for (int t = 0; t < NTILES; ++t) s += p[(size_t)t * NUSE];
    auxsum[idx] = s;
  }
}

// ---------------------------------------------------------------------------
// Aux load-balance loss: sum over segments of n * sum_n (u - 1/n)^2,
// with the relation segment counted twice (Q and K share prefs)
// ---------------------------------------------------------------------------
__global__ void __launch_bounds__(256)
k_aux(const float* __restrict__ auxsum, float* __restrict__ out) {
  __shared__ float rf[256], rr_[256], rv[256];
  int tid = threadIdx.x;
  float sf = 0.f, sr = 0.f, sv = 0.f;
  for (int n = tid; n < NUSE; n += 256) {
    float u = auxsum[n] * (1.0f / (float)(B_ * S_));
    if (n < NF)            { float d = u - 1.0f / NF; sf += d * d; }
    else if (n < NF + NR)  { float d = u - 1.0f / NR; sr += d * d; }
    else                   { float d = u - 1.0f / NV; sv += d * d; }
  }
  rf[tid] = sf; rr_[tid] = sr; rv[tid] = sv; __syncthreads();
  for (int s = 128; s > 0; s >>= 1) {
    if (tid < s) { rf[tid] += rf[tid+s]; rr_[tid] += rr_[tid+s]; rv[tid] += rv[tid+s]; }
    __syncthreads();
  }
  if (tid == 0)
    out[45056] = rf[0] * (float)NF + 2.0f * rr_[0] * (float)NR + rv[0] * (float)NV;
}

// ---------------------------------------------------------------------------
// Top-k scatter + renormalize. k<=8: iterative arg-max with smallest-index
// tiebreak (matches jax.lax.top_k stable ordering). Writes every output slot.
// ---------------------------------------------------------------------------
__global__ void __launch_bounds__(256)
k_topk(const float* __restrict__ dense, float* __restrict__ out) {
  __shared__ float rv[256]; __shared__ int ri[256];
  __shared__ int   chosen[8]; __shared__ float chosenV[8]; __shared__ float sinv;
  int ob = blockIdx.x;
  int b = ob & 7, which = ob >> 3;
  int denseOff, segN, K, outBase;
  if      (which == 0) { denseOff = 0;       segN = NF; K = 8; outBase = 0     + b * NF; }
  else if (which == 1) { denseOff = NF;      segN = NR; K = 4; outBase = 16384 + b * NR; }
  else if (which == 2) { denseOff = NF;      segN = NR; K = 4; outBase = 24576 + b * NR; }
  else                 { denseOff = NF + NR; segN = NV; K = 6; outBase = 32768 + b * NV; }
  const float* row = dense + (size_t)b * NUSE + denseOff;
  int tid = threadIdx.x;

  for (int it = 0; it < K; ++it) {
    float bv = -INFINITY; int bi = -1;
    for (int n = tid; n < segN; n += 256) {
      bool taken = false;
      for (int j = 0; j < it; ++j) taken |= (chosen[j] == n);
      if (taken) continue;
      float v = row[n];
      if (v > bv || (v == bv && (bi < 0 || n < bi))) { bv = v; bi = n; }
    }
    rv[tid] = bv; ri[tid] = bi; __syncthreads();
    for (int s = 128; s > 0; s >>= 1) {
      if (tid < s) {
        float ov = rv[tid + s]; int oi = ri[tid + s];
        if (ov > rv[tid] || (ov == rv[tid] && oi >= 0 && (ri[tid] < 0 || oi < ri[tid]))) {
          rv[tid] = ov; ri[tid] = oi;
        }
      }
      __syncthreads();
    }
    if (tid == 0) { chosen[it] = ri[0]; chosenV[it] = rv[0]; }
    __syncthreads();
  }
  if (tid == 0) {
    float s = 0.f;
    for (int j = 0; j < K; ++j) s += chosenV[j];
    sinv = 1.0f / (s + 1e-8f);
  }
  __syncthreads();
  for (int n = tid; n < segN; n += 256) {
    float v = 0.f;
    for (int j = 0; j < K; ++j) if (chosen[j] == n) v = chosenV[j] * sinv;
    out[outBase + n] = v;
  }
}

// ---------------------------------------------------------------------------
extern "C" void kernel_launch(void* const* d_in, const int* in_sizes, int n_in,
                              void* d_out, int out_size, void* d_ws, size_t ws_size,
                              hipStream_t stream) {
  (void)in_sizes; (void)n_in; (void)out_size; (void)ws_size;
  const float* x          = (const float*)d_in[0];
  const float* importance = (const float*)d_in[1];
  const float* W          = (const float*)d_in[2];
  const float* bproj      = (const float*)d_in[3];
  const float* emb        = (const float*)d_in[4];
  const float* uf         = (const float*)d_in[5];
  const float* ur         = (const float*)d_in[6];
  const float* uv         = (const float*)d_in[7];
  float* out = (float*)d_out;
  float* ws  = (float*)d_ws;

  size_t off = 0;
  float* embn   = ws + off; off += (size_t)NUSE * 64;      // 1.2 MB
  float* bias   = ws + off; off += NUSE;
  float* Wpk    = ws + off; off += (size_t)D_ * 64;        // 0.5 MB pair-packed W
  float* h      = ws + off; off += (size_t)ROWS * 64;      // 4 MB
  float* partA  = ws + off; off += (size_t)NTILES * NUSE;  // 18.9 MB
  float* partB  = ws + off; off += (size_t)NTILES * NUSE;  // 18.9 MB
  float* auxsum = ws + off; off += NUSE;
  float* dense  = ws + off; off += (size_t)B_ * NUSE;      // total ~43.8 MB

  k_prep<<<(NUSE + 255) / 256, 256, 0, stream>>>(emb, uf, ur, uv, embn, bias);
  k_packW<<<((D_ / 2) * 64 + 255) / 256, 256, 0, stream>>>(W, Wpk);
  k_proj<<<ROWS / 64, 128, 0, stream>>>(x, Wpk, bproj, h);
  dim3 gB(NTILES, 3);
  k_logits<<<gB, 256, 0, stream>>>(h, embn, bias, importance, partA, partB);
  k_reduce<<<(B_ * NUSE + 255) / 256, 256, 0, stream>>>(partA, partB, auxsum, dense);
  k_aux<<<1, 256, 0, stream>>>(auxsum, out);
  k_topk<<<32, 256, 0, stream>>>(dense, out);
}